// DRM_Matching_80650895884812
// MI455X (gfx1250) — compile-verified
//
#include <hip/hip_runtime.h>

typedef __attribute__((ext_vector_type(2))) float v2f;
typedef __attribute__((ext_vector_type(8))) float v8f;

#define B_ 32
#define H_ 50
#define L_ 128
#define D_ 256
#define K_ 16
#define EPS_ 1e-12f

// One workgroup (8 wave32s) per (b,h).
//  Phase 1: scores via V_WMMA_F32_16X16X4_F32 (A = sel 16x4 tile, B = user bcast)
//           + per-row self-norm on the VALU (co-executes with WMMA).
//  Phase 2: finalize cosine scores.
//  Phase 3: exact stable top-16 by rank counting (matches lax.top_k ties).
//  Phase 4: coalesced float4 gather of news rows, scaled by thresholded score.
__global__ __launch_bounds__(256) void DRM_Matching_kernel(
    const float* __restrict__ sel,     // [B,H,L,D]
    const float* __restrict__ news,    // [B,H,L,D]
    const float* __restrict__ user,    // [B,1,D]
    const float* __restrict__ thr_p,   // device scalar
    float* __restrict__ out_w,         // [B,H,K,D]
    int*   __restrict__ out_idx)       // [B,H,K]
{
    __shared__ float s_user[D_];
    __shared__ float s_dots[L_];
    __shared__ float s_norm[L_];
    __shared__ float s_scores[L_];
    __shared__ float s_w[K_];
    __shared__ int   s_idx[K_];
    __shared__ float s_uscale;

    const int bh   = blockIdx.x;       // 0 .. B*H-1
    const int b    = bh / H_;
    const int t    = threadIdx.x;      // 0 .. 255
    const int wv   = t >> 5;           // wave 0..7 (wave32)
    const int lane = t & 31;
    const int m    = lane & 15;        // row within 16-row tile
    const int hi   = lane >> 4;        // lane half: K-offset 0 or 2

    const float threshold = thr_p[0];
    const float* selbh  = sel  + (size_t)bh * (L_ * D_);
    const float* newsbh = news + (size_t)bh * (L_ * D_);

    // Stage user vector for this batch into LDS.
    s_user[t] = user[b * D_ + t];
    __syncthreads();

    // Wave 0: 1 / max(||u||, eps)
    if (t < 32) {
        float s = 0.f;
        for (int i = t; i < D_; i += 32) { float v = s_user[i]; s += v * v; }
        for (int off = 16; off > 0; off >>= 1) s += __shfl_xor(s, off, 32);
        if (t == 0) s_uscale = 1.0f / fmaxf(sqrtf(s), EPS_);
    }

    // ---- Phase 1: dots via f32 WMMA, norms via co-executing VALU ----
    const int m0 = wv * 16;
    const float* arow = selbh + (size_t)(m0 + m) * D_;
    v8f  c   = {};
    float nrm = 0.f;
    #pragma unroll 8
    for (int k = 0; k < D_; k += 4) {
        const int kk = k + 2 * hi;
        // A 16x4 fragment: lane m -> (row m, K=k..k+1); lane m+16 -> K=k+2..k+3
        v2f a = *(const v2f*)(arow + kk);
        // B 4x16 fragment: user value broadcast across all 16 columns
        v2f bf; bf[0] = s_user[kk]; bf[1] = s_user[kk + 1];
        c = __builtin_amdgcn_wmma_f32_16x16x4_f32(
                false, a, false, bf, (short)0, c, false, false);
        nrm = fmaf(a[0], a[0], nrm);
        nrm = fmaf(a[1], a[1], nrm);
    }
    // row-m norm^2 lives split across lane m and lane m+16
    nrm += __shfl_xor(nrm, 16, 32);
    if (lane < 16) s_norm[m0 + m] = nrm;
    // All 16 D-columns are identical (B columns were broadcast):
    // lane 0 holds rows m0..m0+7, lane 16 holds rows m0+8..m0+15 (VGPR j -> row j).
    if (m == 0) {
        #pragma unroll
        for (int j = 0; j < 8; ++j) s_dots[m0 + 8 * hi + j] = c[j];
    }
    __syncthreads();

    // ---- Phase 2: cosine scores ----
    if (t < L_) {
        s_scores[t] = s_dots[t] * s_uscale / fmaxf(sqrtf(s_norm[t]), EPS_);
    }
    __syncthreads();

    // ---- Phase 3: exact top-16, lax.top_k tie semantics ----
    if (t < L_) {
        const float st = s_scores[t];
        int r = 0;
        for (int j = 0; j < L_; ++j) {
            const float sj = s_scores[j];
            r += (sj > st) || (sj == st && j < t);
        }
        if (r < K_) {
            s_idx[r] = t;
            s_w[r]   = (st < threshold) ? 0.0f : st;
            out_idx[bh * K_ + r] = t;
        }
    }
    __syncthreads();

    // ---- Phase 4: gather + scale, coalesced 16B accesses ----
    float* obh = out_w + (size_t)bh * (K_ * D_);
    for (int chunk = t; chunk < K_ * (D_ / 4); chunk += 256) {
        const int   r   = chunk >> 6;          // 64 float4 per row
        const int   cc  = chunk & 63;
        const int   idx = s_idx[r];
        const float w   = s_w[r];
        float4 v = *(const float4*)(newsbh + (size_t)idx * D_ + cc * 4);
        float4 o; o.x = v.x * w; o.y = v.y * w; o.z = v.z * w; o.w = v.w * w;
        *(float4*)(obh + (size_t)r * D_ + cc * 4) = o;
    }
}

extern "C" void kernel_launch(void* const* d_in, const int* in_sizes, int n_in,
                              void* d_out, int out_size, void* d_ws, size_t ws_size,
                              hipStream_t stream) {
    const float* sel   = (const float*)d_in[0];  // news_selection_embedding
    const float* news  = (const float*)d_in[1];  // news_embedding
    const float* user  = (const float*)d_in[2];  // user_repr
    // d_in[3] = k (compile-time K_=16), d_in[4] = threshold (device scalar)
    const float* thr_p = (const float*)d_in[4];

    float* out_w   = (float*)d_out;
    int*   out_idx = (int*)((float*)d_out + (size_t)B_ * H_ * K_ * D_);

    dim3 grid(B_ * H_);   // 1600 workgroups
    dim3 block(256);      // 8 wave32s
    DRM_Matching_kernel<<<grid, block, 0, stream>>>(
        sel, news, user, thr_p, out_w, out_idx);
}